// SequenceSelection_7086696038603
// MI455X (gfx1250) — compile-verified
//
#include <hip/hip_runtime.h>
#include <hip/hip_bf16.h>

// Problem constants (match reference)
#define BB 4
#define DD 128
#define EE 256
#define CC 512

typedef float v2f __attribute__((ext_vector_type(2)));
typedef float v8f __attribute__((ext_vector_type(8)));

// log(float32(1e-45)) : 1e-45f rounds to the smallest denormal 1.40129846e-45,
// ln of that is -103.278836. This matches jnp.log(mask + 1e-45) for mask==0.
#define LOG_MASK_EPS (-103.278836f)

__device__ __forceinline__ float dev_tanh(float x) {
#if __has_builtin(__builtin_amdgcn_tanhf)
    return __builtin_amdgcn_tanhf(x);          // v_tanh_f32 (gfx1250 TRANS op)
#elif __has_builtin(__builtin_amdgcn_tanh_f32)
    return __builtin_amdgcn_tanh_f32(x);
#else
    return tanhf(x);
#endif
}

// ---------------------------------------------------------------------------
// Kernel 1: Y[M][512] = X[M][512] @ W^T, W is [512][512] row-major (out, in).
// One wave (32 threads) computes one 16x16 tile of Y with V_WMMA_F32_16X16X4_F32.
//
// ISA §7.12.2 layouts (wave32):
//   A 16x4 f32 : lane m=lane&15 holds row M=m; h=lane>>4 selects K-half;
//                a[v] = A[m][2h + v]           (v = VGPR index 0..1)
//   B 4x16 f32 : lane provides N=lane&15;  b[v] = B[2h + v][n]
//   C/D 16x16  : c[r] = C[r + 8h][lane&15]
// Since both X and W are row-major with K contiguous, A and B fragment loads
// are identical aligned float2 loads from the respective rows.
// ---------------------------------------------------------------------------
__global__ __launch_bounds__(32) void wmma_proj_kernel(
    const float* __restrict__ X,   // [M, 512]
    const float* __restrict__ W,   // [512, 512]
    float* __restrict__ Y)         // [M, 512]
{
    const int lane = threadIdx.x;        // 0..31
    const int m    = lane & 15;
    const int h    = lane >> 4;
    const int m0   = blockIdx.x * 16;    // output row tile
    const int n0   = blockIdx.y * 16;    // output col tile

    const float* xr = X + (size_t)(m0 + m) * CC + 2 * h;  // A row, K-half offset
    const float* wr = W + (size_t)(n0 + m) * CC + 2 * h;  // B "row" (W row n0+m)

    v8f acc = {};
#pragma unroll 4
    for (int k0 = 0; k0 < CC; k0 += 4) {
        v2f a = *(const v2f*)(xr + k0);   // 8B aligned: k0 even, 2h even
        v2f b = *(const v2f*)(wr + k0);
        acc = __builtin_amdgcn_wmma_f32_16x16x4_f32(
            /*neg_a=*/false, a, /*neg_b=*/false, b,
            /*c_mod=*/(short)0, acc, /*reuse_a=*/false, /*reuse_b=*/false);
    }

    float* yr = Y + n0 + m;
#pragma unroll
    for (int r = 0; r < 8; ++r)
        yr[(size_t)(m0 + r + 8 * h) * CC] = acc[r];
}

// ---------------------------------------------------------------------------
// Kernel 2: fused  weights[b,d,e] = sum_c tanh(dec_proj[b,d,c]+enc_proj[b,e,c])*Wde[c]
//           + masked log-softmax over e.
// One block (256 threads, 8 waves) per (b,d). dec row + Wde staged in LDS.
// Each wave owns 32 consecutive e rows; its 32 lanes split C=512 into 16
// contiguous floats each (float4 loads -> the wave streams 2KB contiguously),
// then a shfl_xor tree reduces the dot product.
// ---------------------------------------------------------------------------
__global__ __launch_bounds__(256) void attn_fused_kernel(
    const float* __restrict__ dec_proj,  // [B, D, C]
    const float* __restrict__ enc_proj,  // [B, E, C]
    const int*   __restrict__ mask,      // [B, D, E]
    const float* __restrict__ Wde,       // [C]
    float* __restrict__ out)             // [B, D, E]
{
    __shared__ float s_dec[CC];
    __shared__ float s_wde[CC];
    __shared__ float s_logit[EE];
    __shared__ float s_red[8];

    const int bd   = blockIdx.x;          // 0..B*D-1
    const int b    = bd / DD;
    const int tid  = threadIdx.x;         // 0..255
    const int wave = tid >> 5;
    const int lane = tid & 31;

    // Stage decoder row and Wde into LDS (coalesced).
    const float* decr = dec_proj + (size_t)bd * CC;
    s_dec[tid]       = decr[tid];
    s_dec[tid + 256] = decr[tid + 256];
    s_wde[tid]       = Wde[tid];
    s_wde[tid + 256] = Wde[tid + 256];
    __syncthreads();

    const float* encb  = enc_proj + (size_t)b * EE * CC;
    const int    cbase = lane * 16;       // each lane covers 16 consecutive c

    for (int ei = 0; ei < 32; ++ei) {
        const int e = wave * 32 + ei;
        const float* encr = encb + (size_t)e * CC + cbase;
        float acc = 0.f;
#pragma unroll
        for (int j = 0; j < 4; ++j) {
            const float4 ev = *(const float4*)(encr + 4 * j);
            const int c0 = cbase + 4 * j;
            acc += dev_tanh(ev.x + s_dec[c0 + 0]) * s_wde[c0 + 0];
            acc += dev_tanh(ev.y + s_dec[c0 + 1]) * s_wde[c0 + 1];
            acc += dev_tanh(ev.z + s_dec[c0 + 2]) * s_wde[c0 + 2];
            acc += dev_tanh(ev.w + s_dec[c0 + 3]) * s_wde[c0 + 3];
        }
        // wave32 tree reduction
#pragma unroll
        for (int off = 16; off > 0; off >>= 1)
            acc += __shfl_xor(acc, off, 32);
        if (lane == 0) s_logit[e] = acc;
    }
    __syncthreads();

    // Masked log-softmax over E=256; thread tid owns e = tid.
    const int* mrow = mask + (size_t)bd * EE;
    const float lg = s_logit[tid] + (mrow[tid] ? 0.f : LOG_MASK_EPS);

    // block max
    float v = lg;
#pragma unroll
    for (int off = 16; off > 0; off >>= 1)
        v = fmaxf(v, __shfl_xor(v, off, 32));
    if (lane == 0) s_red[wave] = v;
    __syncthreads();
    float mx = s_red[0];
#pragma unroll
    for (int w = 1; w < 8; ++w) mx = fmaxf(mx, s_red[w]);
    __syncthreads();

    // block sum of exp
    float ex = __expf(lg - mx);
    float s = ex;
#pragma unroll
    for (int off = 16; off > 0; off >>= 1)
        s += __shfl_xor(s, off, 32);
    if (lane == 0) s_red[wave] = s;
    __syncthreads();
    float tot = 0.f;
#pragma unroll
    for (int w = 0; w < 8; ++w) tot += s_red[w];

    out[(size_t)bd * EE + tid] = lg - mx - __logf(tot);
}

// ---------------------------------------------------------------------------
extern "C" void kernel_launch(void* const* d_in, const int* in_sizes, int n_in,
                              void* d_out, int out_size, void* d_ws, size_t ws_size,
                              hipStream_t stream) {
    (void)in_sizes; (void)n_in; (void)out_size; (void)ws_size;

    const float* dec  = (const float*)d_in[0];  // [B,D,C]
    const float* enc  = (const float*)d_in[1];  // [B,E,C]
    const int*   mask = (const int*)d_in[2];    // [B,D,E]
    const float* We   = (const float*)d_in[3];  // [C,C]
    const float* Wd   = (const float*)d_in[4];  // [C,C]
    const float* Wde  = (const float*)d_in[5];  // [C]
    float* out = (float*)d_out;

    // Workspace: enc_proj (B*E*C floats = 2MB) then dec_proj (B*D*C = 1MB).
    float* enc_proj = (float*)d_ws;
    float* dec_proj = enc_proj + (size_t)BB * EE * CC;

    // enc_proj = enc @ We^T   (M = B*E = 1024)
    wmma_proj_kernel<<<dim3((BB * EE) / 16, CC / 16), dim3(32), 0, stream>>>(
        enc, We, enc_proj);
    // dec_proj = dec @ Wd^T   (M = B*D = 512)
    wmma_proj_kernel<<<dim3((BB * DD) / 16, CC / 16), dim3(32), 0, stream>>>(
        dec, Wd, dec_proj);

    // Fused tanh-dot + masked log-softmax: one block per (b,d).
    attn_fused_kernel<<<dim3(BB * DD), dim3(256), 0, stream>>>(
        dec_proj, enc_proj, mask, Wde, out);
}